// RelationScorer_13632226198204
// MI455X (gfx1250) — compile-verified
//
#include <hip/hip_runtime.h>
#include <hip/hip_bf16.h>
#include <math.h>

typedef __attribute__((ext_vector_type(2))) float v2f;
typedef __attribute__((ext_vector_type(8))) float v8f;

#define BB 8
#define NN 400
#define DD 768
#define MM 160
#define PP (MM * (MM - 1))   // 25440

// ---------------------------------------------------------------------------
// Kernel 1: h[b,n] = sigmoid(x[b,n,:] . w_span + b_span)   (one wave per row)
// ---------------------------------------------------------------------------
__global__ void span_h_kernel(const float* __restrict__ x,
                              const float* __restrict__ w_span,
                              const float* __restrict__ b_span,
                              float* __restrict__ h) {
    const int wave = (blockIdx.x * blockDim.x + threadIdx.x) >> 5;  // 0..3199
    const int lane = threadIdx.x & 31;
    const float* row = x + (size_t)wave * DD;
    float s = 0.f;
    #pragma unroll 4
    for (int t = lane; t < DD; t += 32) s += row[t] * w_span[t];
    #pragma unroll
    for (int off = 16; off > 0; off >>= 1) s += __shfl_down(s, off, 32);
    if (lane == 0) h[wave] = 1.f / (1.f + expf(-(s + b_span[0])));
}

// ---------------------------------------------------------------------------
// Kernel 2: per batch: rank(i) for i<160 (stable descending), pos = sorted ranks
// ---------------------------------------------------------------------------
__global__ void rank_pos_kernel(const float* __restrict__ h, int* __restrict__ pos) {
    __shared__ float hs[NN];
    __shared__ int   rs[MM];
    const int b = blockIdx.x;
    for (int t = threadIdx.x; t < NN; t += blockDim.x) hs[t] = h[b * NN + t];
    __syncthreads();
    for (int i = threadIdx.x; i < MM; i += blockDim.x) {
        const float hi = hs[i];
        int cnt = 0;
        for (int j = 0; j < NN; ++j) {
            const float hj = hs[j];
            cnt += (hj > hi) || ((hj == hi) && (j < i));
        }
        rs[i] = cnt;
    }
    __syncthreads();
    for (int i = threadIdx.x; i < MM; i += blockDim.x) {
        const int r = rs[i];
        int idx = 0;
        for (int i2 = 0; i2 < MM; ++i2) idx += (rs[i2] < r);
        pos[b * MM + idx] = r;
    }
}

// ---------------------------------------------------------------------------
// Kernel 3: gather x_ranked rows + ranges_ranked (one block per (b,i))
// ---------------------------------------------------------------------------
__global__ void gather_kernel(const float* __restrict__ x,
                              const int* __restrict__ pos,
                              const int* __restrict__ span_ranges,
                              float* __restrict__ xr,
                              int* __restrict__ rr) {
    const int row = blockIdx.x;          // b*160 + i
    const int b   = row / MM;
    const int p   = pos[row];
    const float* src = x + ((size_t)b * NN + p) * DD;
    float* dst = xr + (size_t)row * DD;
    for (int t = threadIdx.x; t < DD; t += blockDim.x) dst[t] = src[t];
    if (threadIdx.x == 0) {
        rr[row * 2 + 0] = span_ranges[p * 2 + 0];
        rr[row * 2 + 1] = span_ranges[p * 2 + 1];
    }
}

// ---------------------------------------------------------------------------
// Kernel 4: transpose w3 slice of w_pair -> w3t[c*768 + d]
// ---------------------------------------------------------------------------
__global__ void w3t_kernel(const float* __restrict__ w_pair, float* __restrict__ w3t) {
    for (int t = blockIdx.x * blockDim.x + threadIdx.x; t < 3 * DD;
         t += gridDim.x * blockDim.x) {
        const int c = t / DD, d = t % DD;
        w3t[t] = w_pair[(size_t)(2 * DD + d) * 3 + c];
    }
}

// ---------------------------------------------------------------------------
// Kernel 5: avec = x_r @ w1, bvec = x_r @ w2   (one wave per (b,i))
// ---------------------------------------------------------------------------
__global__ void ab_kernel(const float* __restrict__ xr,
                          const float* __restrict__ w_pair,
                          float* __restrict__ avec,
                          float* __restrict__ bvec) {
    const int wave = (blockIdx.x * blockDim.x + threadIdx.x) >> 5;  // b*160+i
    const int lane = threadIdx.x & 31;
    const float* row = xr + (size_t)wave * DD;
    float a0 = 0.f, a1 = 0.f, a2 = 0.f, c0 = 0.f, c1 = 0.f, c2 = 0.f;
    for (int d = lane; d < DD; d += 32) {
        const float xv = row[d];
        a0 += xv * w_pair[d * 3 + 0];
        a1 += xv * w_pair[d * 3 + 1];
        a2 += xv * w_pair[d * 3 + 2];
        c0 += xv * w_pair[(DD + d) * 3 + 0];
        c1 += xv * w_pair[(DD + d) * 3 + 1];
        c2 += xv * w_pair[(DD + d) * 3 + 2];
    }
    #pragma unroll
    for (int off = 16; off > 0; off >>= 1) {
        a0 += __shfl_xor(a0, off, 32); a1 += __shfl_xor(a1, off, 32);
        a2 += __shfl_xor(a2, off, 32); c0 += __shfl_xor(c0, off, 32);
        c1 += __shfl_xor(c1, off, 32); c2 += __shfl_xor(c2, off, 32);
    }
    if (lane == 0) {
        avec[wave * 3 + 0] = a0; avec[wave * 3 + 1] = a1; avec[wave * 3 + 2] = a2;
        bvec[wave * 3 + 0] = c0; bvec[wave * 3 + 1] = c1; bvec[wave * 3 + 2] = c2;
    }
}

// ---------------------------------------------------------------------------
// Kernel 6: main WMMA kernel. One block per (b, i_tile); wave w owns j_tile=w.
// Per k-step: A_c = X[i-rows]*w3[:,c]  (fp32 16x16x4 WMMA), B = X[j-rows]^T.
// Epilogue: logits -> sigmoid -> softmax(C=3) -> scatter off-diagonal pairs.
// ---------------------------------------------------------------------------
__global__ void __launch_bounds__(320)
gram_probs_kernel(const float* __restrict__ xr,
                  const float* __restrict__ w3t,
                  const float* __restrict__ avec,
                  const float* __restrict__ bvec,
                  const float* __restrict__ b_pair,
                  float* __restrict__ out_probs) {
    const int b    = blockIdx.x / 10;
    const int i0   = (blockIdx.x % 10) * 16;
    const int wave = threadIdx.x >> 5;        // j_tile 0..9
    const int lane = threadIdx.x & 31;
    const int half = lane >> 4;
    const int mn   = lane & 15;
    const int j0   = wave * 16;

    const float* xb   = xr + (size_t)b * MM * DD;
    const float* rowA = xb + (size_t)(i0 + mn) * DD;   // A: row i0+m, cols k
    const float* rowB = xb + (size_t)(j0 + mn) * DD;   // B: col j0+n = row of X
    const float* w0p  = w3t + 0 * DD;
    const float* w1p  = w3t + 1 * DD;
    const float* w2p  = w3t + 2 * DD;

    v8f acc0 = {}; v8f acc1 = {}; v8f acc2 = {};
    for (int k0 = 0; k0 < DD; k0 += 4) {
        const int kk = k0 + 2 * half;                  // K = 2*half + vgpr
        const v2f a2 = *(const v2f*)(rowA + kk);
        const v2f b2 = *(const v2f*)(rowB + kk);
        const v2f w0 = *(const v2f*)(w0p + kk);
        const v2f w1 = *(const v2f*)(w1p + kk);
        const v2f w2 = *(const v2f*)(w2p + kk);
        const v2f A0 = a2 * w0;
        const v2f A1 = a2 * w1;
        const v2f A2 = a2 * w2;
        acc0 = __builtin_amdgcn_wmma_f32_16x16x4_f32(false, A0, false, b2,
                                                     (short)0, acc0, false, false);
        acc1 = __builtin_amdgcn_wmma_f32_16x16x4_f32(false, A1, false, b2,
                                                     (short)0, acc1, false, false);
        acc2 = __builtin_amdgcn_wmma_f32_16x16x4_f32(false, A2, false, b2,
                                                     (short)0, acc2, false, false);
    }

    // Epilogue: C/D layout — element (m = v + 8*half, n = lane&15)
    const int j = j0 + mn;
    const float bv0 = bvec[((size_t)b * MM + j) * 3 + 0];
    const float bv1 = bvec[((size_t)b * MM + j) * 3 + 1];
    const float bv2 = bvec[((size_t)b * MM + j) * 3 + 2];
    const float bp0 = b_pair[0], bp1 = b_pair[1], bp2 = b_pair[2];

    #pragma unroll
    for (int v = 0; v < 8; ++v) {
        const int m = v + 8 * half;
        const int i = i0 + m;
        if (i == j) continue;
        const float l0 = acc0[v] + avec[((size_t)b * MM + i) * 3 + 0] + bv0 + bp0;
        const float l1 = acc1[v] + avec[((size_t)b * MM + i) * 3 + 1] + bv1 + bp1;
        const float l2 = acc2[v] + avec[((size_t)b * MM + i) * 3 + 2] + bv2 + bp2;
        const float s0 = 1.f / (1.f + expf(-l0));
        const float s1 = 1.f / (1.f + expf(-l1));
        const float s2 = 1.f / (1.f + expf(-l2));
        const float mx = fmaxf(s0, fmaxf(s1, s2));
        const float e0 = expf(s0 - mx), e1 = expf(s1 - mx), e2 = expf(s2 - mx);
        const float inv = 1.f / (e0 + e1 + e2);
        const size_t p = (size_t)i * 159 + j - (j > i ? 1 : 0);
        float* o = out_probs + ((size_t)b * PP + p) * 3;
        o[0] = e0 * inv; o[1] = e1 * inv; o[2] = e2 * inv;
    }
}

// ---------------------------------------------------------------------------
// Kernel 7: pair_ranges: (B,P,2,2) as floats
// ---------------------------------------------------------------------------
__global__ void ranges_kernel(const int* __restrict__ rr, float* __restrict__ out2) {
    const int t = blockIdx.x * blockDim.x + threadIdx.x;
    if (t >= BB * PP) return;
    const int b = t / PP, p = t % PP;
    const int i = p / 159, r = p % 159;
    const int j = r + (r >= i ? 1 : 0);
    float* o = out2 + (size_t)t * 4;
    o[0] = (float)rr[(b * MM + i) * 2 + 0];
    o[1] = (float)rr[(b * MM + i) * 2 + 1];
    o[2] = (float)rr[(b * MM + j) * 2 + 0];
    o[3] = (float)rr[(b * MM + j) * 2 + 1];
}

extern "C" void kernel_launch(void* const* d_in, const int* in_sizes, int n_in,
                              void* d_out, int out_size, void* d_ws, size_t ws_size,
                              hipStream_t stream) {
    const float* x           = (const float*)d_in[0];   // (B,N,D)
    const float* w_span      = (const float*)d_in[1];   // (D,1)
    const float* b_span      = (const float*)d_in[2];   // (1,)
    const float* w_pair      = (const float*)d_in[3];   // (3D,3)
    const float* b_pair      = (const float*)d_in[4];   // (3,)
    const int*   span_ranges = (const int*)d_in[5];     // (N,2)

    float* out_probs  = (float*)d_out;                        // B*P*3
    float* out_ranges = (float*)d_out + (size_t)BB * PP * 3;  // B*P*2*2

    // workspace layout (floats)
    float* ws   = (float*)d_ws;
    size_t off  = 0;
    float* h    = ws + off;            off += (size_t)BB * NN;       // 3200
    int*   pos  = (int*)(ws + off);    off += (size_t)BB * MM;       // 1280
    float* xr   = ws + off;            off += (size_t)BB * MM * DD;  // 983040
    float* avec = ws + off;            off += (size_t)BB * MM * 3;
    float* bvec = ws + off;            off += (size_t)BB * MM * 3;
    int*   rr   = (int*)(ws + off);    off += (size_t)BB * MM * 2;
    float* w3t  = ws + off;            off += (size_t)3 * DD;

    // 1) h = sigmoid(x . w_span + b_span): 3200 waves, 8 waves/block
    span_h_kernel<<<(BB * NN) / 8, 256, 0, stream>>>(x, w_span, b_span, h);
    // 2) ranks + pos per batch
    rank_pos_kernel<<<BB, 256, 0, stream>>>(h, pos);
    // 3) gather x_ranked + ranges_ranked
    gather_kernel<<<BB * MM, 256, 0, stream>>>(x, pos, span_ranges, xr, rr);
    // 4) transpose w3
    w3t_kernel<<<9, 256, 0, stream>>>(w_pair, w3t);
    // 5) avec/bvec: 1280 waves, 8 waves/block
    ab_kernel<<<(BB * MM) / 8, 256, 0, stream>>>(xr, w_pair, avec, bvec);
    // 6) WMMA Gram + fused sigmoid/softmax/scatter: (b,i_tile) blocks, 10 waves
    gram_probs_kernel<<<BB * 10, 320, 0, stream>>>(xr, w3t, avec, bvec, b_pair,
                                                   out_probs);
    // 7) pair_ranges
    ranges_kernel<<<(BB * PP + 255) / 256, 256, 0, stream>>>(rr, out_ranges);
}